// RoiPool_27479200760160
// MI455X (gfx1250) — compile-verified
//
#include <hip/hip_runtime.h>
#include <stdint.h>

// RoI max pooling (Caffe-style), CDNA5/gfx1250.
// data:[B=4,C=256,H=64,W=64] f32, rois:[N,4] f32, roibatches:[N] i32,
// spatial_scale: scalar f32. out:[N,C,7,7] f32.
//
// One block per (roi, 2-channel chunk). Stage the RoI's [row window] x
// [16B-aligned column window] into LDS with gfx1250 async global->LDS copies
// (ASYNCcnt path), then 98 threads compute the 2*7*7 bin maxima from LDS.
// Memory-bound op; no matmul -> no WMMA; async-tensor path + scalar-FP
// boundary math are the CDNA5-specific wins.

#define CROP_H 7
#define CROP_W 7
#define CPB    2     // channels per block
#define BLK    128   // 4 wave32s

__global__ __launch_bounds__(BLK) void roipool_gfx1250(
    const float* __restrict__ data,
    const float* __restrict__ rois,
    const int*   __restrict__ roibatches,
    const float* __restrict__ scale_ptr,
    float*       __restrict__ out,
    int C, int H, int W)
{
    // Tile: CPB channels x 64 rows x 64 cols (worst case), 32 KB.
    // Row stride is fixed at 64 so the compute phase indexes by absolute w.
    __shared__ float tile[CPB * 64 * 64];
    __shared__ int sh_hs[CROP_H], sh_he[CROP_H];
    __shared__ int sh_ws[CROP_W], sh_we[CROP_W];
    __shared__ int sh_meta[5]; // h_lo, RH, batch, w_al (aligned col base), nseg

    const int r   = blockIdx.x;
    const int c0  = blockIdx.y * CPB;
    const int tid = threadIdx.x;

    if (tid == 0) {
        const float scale = *scale_ptr;
        const float x1 = rois[r * 4 + 0];
        const float y1 = rois[r * 4 + 1];
        const float x2 = rois[r * 4 + 2];
        const float y2 = rois[r * 4 + 3];
        // round-half-even in f32 (s_rndne_f32), matching jnp.round
        const int rsw = (int)rintf(x1 * scale);
        const int rsh = (int)rintf(y1 * scale);
        const int rew = (int)rintf(x2 * scale);
        const int reh = (int)rintf(y2 * scale);
        const float roi_h = (float)max(reh - rsh + 1, 1);
        const float roi_w = (float)max(rew - rsw + 1, 1);
        const float bh = roi_h / (float)CROP_H;
        const float bw = roi_w / (float)CROP_W;
        for (int p = 0; p < CROP_H; ++p) {
            int hs = (int)floorf((float)p * bh) + rsh;
            int he = (int)ceilf(((float)p + 1.0f) * bh) + rsh;
            sh_hs[p] = min(max(hs, 0), H);
            sh_he[p] = min(max(he, 0), H);
        }
        for (int q = 0; q < CROP_W; ++q) {
            int ws = (int)floorf((float)q * bw) + rsw;
            int we = (int)ceilf(((float)q + 1.0f) * bw) + rsw;
            sh_ws[q] = min(max(ws, 0), W);
            sh_we[q] = min(max(we, 0), W);
        }
        const int h_lo = sh_hs[0];                 // bins are monotone
        const int w_al = sh_ws[0] & ~3;            // 16B-aligned column base
        const int w_hi = sh_we[CROP_W - 1];
        sh_meta[0] = h_lo;
        sh_meta[1] = max(sh_he[CROP_H - 1] - h_lo, 0);   // RH rows
        sh_meta[2] = roibatches[r];
        sh_meta[3] = w_al;
        sh_meta[4] = max((w_hi - w_al + 3) >> 2, 0);     // 16B segments per row
    }
    __syncthreads();

    const int h_lo = sh_meta[0];
    const int RH   = sh_meta[1];
    const int b    = sh_meta[2];
    const int w_al = sh_meta[3];
    const int nseg = sh_meta[4];

    // ---- Stage CPB channels x RH rows x nseg 16B-chunks into LDS (async).
    const int per_ch = RH * nseg;
    const int chunks = CPB * per_ch;
    const float* src_base = data + ((size_t)(b * C + c0) * (size_t)(H * W));
    const unsigned lds_base = (unsigned)(uintptr_t)(&tile[0]); // low 32b = LDS offset

    for (int k = tid; k < chunks; k += BLK) {
        const int cl  = k / per_ch;
        const int rem = k - cl * per_ch;
        const int row = rem / nseg;
        const int seg = rem - row * nseg;
        const int col = w_al + (seg << 2);
        const unsigned long long gaddr =
            (unsigned long long)(uintptr_t)(src_base
                + (size_t)cl * (size_t)(H * W)
                + (size_t)(h_lo + row) * (size_t)W
                + (size_t)col);
        const unsigned lds = lds_base
            + (unsigned)((((cl << 6) + row) << 6) + col) * 4u;
        asm volatile("global_load_async_to_lds_b128 %0, %1, off"
                     :: "v"(lds), "v"(gaddr)
                     : "memory");
    }
    // Drain this wave's async copies, then cross-wave barrier.
    asm volatile("s_wait_asynccnt 0" ::: "memory");
    __syncthreads();

    // ---- Compute: thread t -> (channel cl, bin ph,pw).
    if (tid < CPB * CROP_H * CROP_W) {
        const int cl  = tid / (CROP_H * CROP_W);
        const int bin = tid - cl * (CROP_H * CROP_W);
        const int ph  = bin / CROP_W;
        const int pw  = bin - ph * CROP_W;
        const int hs = sh_hs[ph], he = sh_he[ph];
        const int ws = sh_ws[pw], we = sh_we[pw];

        float m = -INFINITY;
        for (int h = hs; h < he; ++h) {
            const float* trow = &tile[(((cl << 6) + (h - h_lo)) << 6)];
            for (int w = ws; w < we; ++w) {
                m = fmaxf(m, trow[w]);
            }
        }
        const float res = (m == -INFINITY) ? 0.0f : m;
        const size_t oidx =
            (((size_t)r * (size_t)C + (size_t)(c0 + cl)) * CROP_H + ph) * CROP_W + pw;
        out[oidx] = res;
    }
}

extern "C" void kernel_launch(void* const* d_in, const int* in_sizes, int n_in,
                              void* d_out, int out_size, void* d_ws, size_t ws_size,
                              hipStream_t stream) {
    const float* data       = (const float*)d_in[0];
    const float* rois       = (const float*)d_in[1];
    const int*   roibatches = (const int*)d_in[2];
    const float* scale_ptr  = (const float*)d_in[3];
    float*       out        = (float*)d_out;

    const int N = in_sizes[1] / 4;   // rois: [N,4]
    const int B = 4, C = 256, H = 64, W = 64;
    (void)B; (void)n_in; (void)d_ws; (void)ws_size; (void)out_size;

    dim3 grid(N, C / CPB);
    roipool_gfx1250<<<grid, BLK, 0, stream>>>(data, rois, roibatches, scale_ptr,
                                              out, C, H, W);
}